// StructureLoss_3994319585839
// MI455X (gfx1250) — compile-verified
//
#include <hip/hip_runtime.h>
#include <cstdint>
#include <cstddef>

// Problem constants (from reference): B=32, C=1, H=512, W=512, 31x31 box, pad 15.
#define Bn 32
#define Hn 512
#define Wn 512
#define Kw 31
#define Rr 15
#define CHUNK 64   // rows per thread in the vertical pass
#define ROWS 4     // rows per block in the fused horizontal pass (4 | 512)

typedef unsigned int v4u_t __attribute__((ext_vector_type(4)));
typedef int          v8i_t __attribute__((ext_vector_type(8)));
typedef int          v4i_t __attribute__((ext_vector_type(4)));

// CDNA5 Tensor Data Mover path (gfx1250). Fallback: cooperative LDS load.
#if defined(__AMDGCN__) && __has_builtin(__builtin_amdgcn_tensor_load_to_lds)
#define USE_TDM 1
#else
#define USE_TDM 0
#endif

__device__ __forceinline__ float wave_sum32(float v) {
#pragma unroll
  for (int off = 16; off > 0; off >>= 1) v += __shfl_down(v, off, 32);
  return v;
}

#if USE_TDM
// Issue one 1-D TDM load: 512 contiguous f32 from `gaddr` into LDS at `lds_off`.
__device__ __forceinline__ void tdm_load_row(unsigned lds_off, unsigned long long ga) {
  // D# group0: count=1, lds_addr, global_addr[56:0], type=2 ("image")
  v4u_t g0 = { 1u, lds_off, (unsigned)ga,
               ((unsigned)(ga >> 32) & 0x01FFFFFFu) | 0x80000000u };
  // D# group1: data_size=4B (code 2), tensor_dim0=512, tensor_dim1=1,
  //            tile_dim0=512 (tile_dim1/2=0 -> 1D), tensor_dim0_stride=512.
  v8i_t g1 = { 0x00020000, 0x02000000, 0x00010000, 0x02000000,
               0,          512,        0,          0 };
  v4i_t z4 = { 0, 0, 0, 0 };
#if __has_include(<hip/amd_detail/amd_gfx1250_TDM.h>)
  v8i_t z8 = { 0, 0, 0, 0, 0, 0, 0, 0 };
  __builtin_amdgcn_tensor_load_to_lds(g0, g1, z4, z4, z8, 0);
#else
  __builtin_amdgcn_tensor_load_to_lds(g0, g1, z4, z4, 0);
#endif
}
#endif

// ---------------------------------------------------------------------------
// Pass A: vertical 31-tap box sum of target with zero padding.
// s1[b,h,w] = sum_{dh=-15..15} t[b,h+dh,w].
// Thread handles a 64-row column chunk with a running-window sum; loads are
// coalesced across w (lane-contiguous 128B lines). t fits in the 192MB L2, so
// the window re-read costs L2 bandwidth, not HBM.
// ---------------------------------------------------------------------------
__global__ void pass_vbox(const float* __restrict__ t, float* __restrict__ s1) {
  const int gid   = blockIdx.x * blockDim.x + threadIdx.x;
  const int w     = gid & (Wn - 1);
  const int rest  = gid >> 9;                  // W = 512
  const int chunk = rest & ((Hn / CHUNK) - 1); // 8 chunks
  const int b     = rest >> 3;

  const float* tp = t  + ((size_t)b * Hn) * Wn + w;
  float*       sp = s1 + ((size_t)b * Hn) * Wn + w;

  const int h0 = chunk * CHUNK;
  float sum = 0.0f;
  int lo = h0 - Rr; if (lo < 0) lo = 0;
  for (int h = lo; h <= h0 + Rr; ++h) sum += tp[(size_t)h * Wn];

  for (int h = h0; h < h0 + CHUNK; ++h) {
    sp[(size_t)h * Wn] = sum;
    const int hin  = h + Rr + 1;
    const int hout = h - Rr;
    if (hin  <  Hn) sum += tp[(size_t)hin  * Wn];
    if (hout >= 0 ) sum -= tp[(size_t)hout * Wn];
  }
}

// ---------------------------------------------------------------------------
// Pass B: one 512-thread block per 4 consecutive rows of one image.
//  - Wave 0 issues 4 TDM DMAs (in-order per wave), one s_wait_tensorcnt 0.
//  - Horizontal 31-tap sum from zero-padded LDS rows (stride-1: conflict-free).
//  - Fused elementwise: weit, stable BCE-with-logits, sigmoid; v0..v3
//    accumulate across the 4 rows in registers, then ONE reduction per block.
//  - Deterministic block partials -> blocksum[blk][4] (no float atomics).
// ---------------------------------------------------------------------------
__global__ void pass_hbox_fused(const float* __restrict__ x,
                                const float* __restrict__ t,
                                const float* __restrict__ s1,
                                float* __restrict__ blocksum) {
  __shared__ float srow[ROWS][Wn + 2 * Rr];  // per row: [0..14]=0, [15..526]=data, [527..541]=0
  __shared__ float red[16][4];

  const int    blk   = blockIdx.x;            // 0 .. B*H/ROWS-1 (rows never straddle images)
  const int    tid   = threadIdx.x;           // == w, 0..511
  const size_t base0 = (size_t)blk * ROWS * Wn;

  // Zero the 15-entry pads of all 4 rows (64 threads, 2 stores each for j < 15, none for j == 15).
  if (tid < ROWS * 16) {
    const int r = tid >> 4, j = tid & 15;
    if (j < Rr) { srow[r][j] = 0.0f; srow[r][Rr + Wn + j] = 0.0f; }
  }

#if USE_TDM
  if (tid < 32) {  // wave 0 issues all four 2KB DMAs, then one wait
#pragma unroll
    for (int r = 0; r < ROWS; ++r) {
      const unsigned lds_off = (unsigned)(uintptr_t)(&srow[r][Rr]); // low 32b of flat = LDS offset
      const unsigned long long ga =
          (unsigned long long)(uintptr_t)(s1 + base0 + (size_t)r * Wn);
      tdm_load_row(lds_off, ga);
    }
#if __has_builtin(__builtin_amdgcn_s_wait_tensorcnt)
    __builtin_amdgcn_s_wait_tensorcnt(0);
#else
    asm volatile("s_wait_tensorcnt 0x0" ::: "memory");
#endif
  }
#else
#pragma unroll
  for (int r = 0; r < ROWS; ++r) srow[r][Rr + tid] = s1[base0 + (size_t)r * Wn + tid];
#endif
  __syncthreads();

  float v0 = 0.0f, v1 = 0.0f, v2 = 0.0f, v3 = 0.0f;
#pragma unroll
  for (int r = 0; r < ROWS; ++r) {
    // 31-tap horizontal sum from LDS (stride-1 across lanes: conflict-free).
    float s = 0.0f;
#pragma unroll
    for (int i = 0; i < Kw; ++i) s += srow[r][tid + i];

    const size_t idx  = base0 + (size_t)r * Wn + tid;
    const float  avg  = s * (1.0f / (31.0f * 31.0f));
    const float  tv   = t[idx];
    const float  xv   = x[idx];
    const float  weit = 1.0f + 5.0f * fabsf(avg - tv);
    const float  bce  = fmaxf(xv, 0.0f) - xv * tv + log1pf(expf(-fabsf(xv)));
    const float  p    = 1.0f / (1.0f + expf(-xv));

    v0 += weit * bce;
    v1 += weit;
    v2 += p * tv * weit;
    v3 += (p + tv) * weit;
  }

  v0 = wave_sum32(v0); v1 = wave_sum32(v1);
  v2 = wave_sum32(v2); v3 = wave_sum32(v3);

  const int lane = tid & 31, wv = tid >> 5;
  if (lane == 0) { red[wv][0] = v0; red[wv][1] = v1; red[wv][2] = v2; red[wv][3] = v3; }
  __syncthreads();
  if (tid == 0) {
    float S0 = 0, S1 = 0, S2 = 0, S3 = 0;
#pragma unroll
    for (int i = 0; i < 16; ++i) { S0 += red[i][0]; S1 += red[i][1]; S2 += red[i][2]; S3 += red[i][3]; }
    float* r = blocksum + (size_t)blk * 4;
    r[0] = S0; r[1] = S1; r[2] = S2; r[3] = S3;
  }
}

// ---------------------------------------------------------------------------
// Finalize 1: per image, reduce 128 block partials -> wbce + wiou.
// ---------------------------------------------------------------------------
__global__ void finalize1(const float* __restrict__ blocksum, float* __restrict__ bloss) {
  const int b = blockIdx.x;
  const int tid = threadIdx.x;  // 256 threads
  const int nblk = Hn / ROWS;   // 128 partials per image
  float v0 = 0, v1 = 0, v2 = 0, v3 = 0;
  for (int i = tid; i < nblk; i += 256) {
    const float* r = blocksum + ((size_t)b * nblk + i) * 4;
    v0 += r[0]; v1 += r[1]; v2 += r[2]; v3 += r[3];
  }
  v0 = wave_sum32(v0); v1 = wave_sum32(v1);
  v2 = wave_sum32(v2); v3 = wave_sum32(v3);
  __shared__ float red[8][4];
  const int lane = tid & 31, wv = tid >> 5;
  if (lane == 0) { red[wv][0] = v0; red[wv][1] = v1; red[wv][2] = v2; red[wv][3] = v3; }
  __syncthreads();
  if (tid == 0) {
    float S0 = 0, S1 = 0, S2 = 0, S3 = 0;
#pragma unroll
    for (int i = 0; i < 8; ++i) { S0 += red[i][0]; S1 += red[i][1]; S2 += red[i][2]; S3 += red[i][3]; }
    const float wbce = S0 / S1;
    const float wiou = 1.0f - (S2 + 1.0f) / (S3 - S2 + 1.0f);
    bloss[b] = wbce + wiou;
  }
}

// Finalize 2: one wave32 == 32 images; mean -> scalar output.
__global__ void finalize2(const float* __restrict__ bloss, float* __restrict__ out) {
  float v = bloss[threadIdx.x];
  v = wave_sum32(v);
  if (threadIdx.x == 0) out[0] = v * (1.0f / 32.0f);
}

// ---------------------------------------------------------------------------
extern "C" void kernel_launch(void* const* d_in, const int* in_sizes, int n_in,
                              void* d_out, int out_size, void* d_ws, size_t ws_size,
                              hipStream_t stream) {
  (void)in_sizes; (void)n_in; (void)out_size; (void)ws_size;
  const float* x = (const float*)d_in[0];   // input (logits)
  const float* t = (const float*)d_in[1];   // target

  float* s1       = (float*)d_ws;                      // 32*512*512 floats (32 MiB)
  float* blocksum = s1 + (size_t)Bn * Hn * Wn;         // (B*H/ROWS)*4 floats
  float* bloss    = blocksum + (size_t)Bn * (Hn / ROWS) * 4; // 32 floats
  float* out      = (float*)d_out;                     // 1 float

  pass_vbox      <<<(Bn * (Hn / CHUNK) * Wn) / 256, 256, 0, stream>>>(t, s1);
  pass_hbox_fused<<<Bn * Hn / ROWS, Wn, 0, stream>>>(x, t, s1, blocksum);
  finalize1      <<<Bn, 256, 0, stream>>>(blocksum, bloss);
  finalize2      <<<1, 32, 0, stream>>>(bloss, out);
}